// SparseMoE_8074538516586
// MI455X (gfx1250) — compile-verified
//
#include <hip/hip_runtime.h>
#include <hip/hip_bf16.h>
#include <math.h>

// ---------------------------------------------------------------------------
// SparseMoE on MI455X (gfx1250).  GEMMs on V_WMMA_F32_16X16X4_F32 (fp32
// matrix path), tiles staged into LDS via GLOBAL_LOAD_ASYNC_TO_LDS_B128
// (ASYNCcnt) when the builtin is available, VALU epilogues, routing/gating
// in small kernels.  Dense-expert formulation (matches reference exactly).
// ---------------------------------------------------------------------------

#define NTOK   8192      // B*T
#define CDIM   768
#define EEXP   6
#define HID    3072      // 4C
#define HALF_H 1536      // 2C

typedef float v2f __attribute__((ext_vector_type(2)));
typedef float v8f __attribute__((ext_vector_type(8)));
typedef int   v4i __attribute__((ext_vector_type(4)));

#define AS1 __attribute__((address_space(1)))
#define AS3 __attribute__((address_space(3)))

#if defined(__has_builtin)
#if __has_builtin(__builtin_amdgcn_global_load_async_to_lds_b128)
#define MOE_ASYNC_LDS 1
#endif
#endif

// stage one float4 global -> LDS
__device__ __forceinline__ void stage_f4(const float* __restrict__ gsrc,
                                         float* __restrict__ ldst) {
#ifdef MOE_ASYNC_LDS
    __builtin_amdgcn_global_load_async_to_lds_b128(
        (AS1 v4i*)gsrc, (AS3 v4i*)ldst, 0, 0);
#else
    float4 v = *(const float4*)gsrc;
    ldst[0] = v.x; ldst[1] = v.y; ldst[2] = v.z; ldst[3] = v.w;
#endif
}

__device__ __forceinline__ void stage_wait() {
#ifdef MOE_ASYNC_LDS
#if __has_builtin(__builtin_amdgcn_s_wait_asynccnt)
    __builtin_amdgcn_s_wait_asynccnt(0);
#else
    asm volatile("s_wait_asynccnt 0x0" ::: "memory");
#endif
#endif
}

__device__ __forceinline__ float gelu_f(float v) {
    return 0.5f * v * (1.0f + erff(v * 0.70710678118654752f));
}
__device__ __forceinline__ float silu_f(float v) {
    return v / (1.0f + expf(-v));
}
__device__ __forceinline__ float act_f(float v, int act) {
    if (act == 1) return gelu_f(v);
    if (act == 2) return silu_f(v);
    return v;
}
__device__ __forceinline__ float softplus_f(float v) {
    return (v > 20.0f) ? v : log1pf(expf(v));
}

// ---------------------------------------------------------------------------
// fp32 WMMA GEMM:  C[M,N] (=|+=) act( A[M,K] @ W[K,N] + bias[N] )
// Block tile 64x64, 128 threads (4 wave32s).  Each wave owns a 16-row slab
// and the full 64 columns: 4 f32 16x16 accumulators, one A fragment reused
// 4x per k-step.  BK = 32 staged through LDS (async-to-LDS when available).
// Requires M%64==0, N%64==0, K%32==0  (M=8192, K,N in {768,1536,3072}).
// ---------------------------------------------------------------------------
__global__ __launch_bounds__(128)
void moe_wmma_gemm(const float* __restrict__ A, const float* __restrict__ W,
                   const float* __restrict__ bias, float* __restrict__ C,
                   int M, int N, int K, int act, int accum) {
    __shared__ float As[64][36];   // 64 x 32 tile of A (row pad: 144B, 16B-aligned)
    __shared__ float Ws[32][68];   // 32 x 64 tile of W (row pad: 272B, 16B-aligned)

    const int tid  = threadIdx.x;
    const int lane = tid & 31;
    const int wave = tid >> 5;          // 0..3 -> 16-row slab
    const int l15  = lane & 15;
    const int hi   = lane >> 4;         // upper half-wave

    const int m0 = blockIdx.y * 64;
    const int n0 = blockIdx.x * 64;

    v8f acc[4];
    acc[0] = {}; acc[1] = {}; acc[2] = {}; acc[3] = {};

    for (int k0 = 0; k0 < K; k0 += 32) {
        // ---- stage A tile: 64x32 = 512 float4 (4 per thread) ----
        #pragma unroll
        for (int i = tid; i < 512; i += 128) {
            int r  = i >> 3;
            int c4 = (i & 7) << 2;
            stage_f4(A + (size_t)(m0 + r) * K + k0 + c4, &As[r][c4]);
        }
        // ---- stage W tile: 32x64 = 512 float4 (4 per thread) ----
        #pragma unroll
        for (int i = tid; i < 512; i += 128) {
            int r  = i >> 4;
            int c4 = (i & 15) << 2;
            stage_f4(W + (size_t)(k0 + r) * N + n0 + c4, &Ws[r][c4]);
        }
        stage_wait();
        __syncthreads();

        // ---- 8 WMMA k-steps of 4; A fragment reused across 4 col tiles ----
        #pragma unroll
        for (int kk = 0; kk < 32; kk += 4) {
            v2f a;
            // A 16x4 frag: lanes 0-15 hold K=kk,kk+1; lanes 16-31 K=kk+2,kk+3
            a.x = As[16 * wave + l15][kk + 2 * hi];
            a.y = As[16 * wave + l15][kk + 2 * hi + 1];
            #pragma unroll
            for (int j = 0; j < 4; ++j) {
                v2f b;
                b.x = Ws[kk + 2 * hi][16 * j + l15];
                b.y = Ws[kk + 2 * hi + 1][16 * j + l15];
                acc[j] = __builtin_amdgcn_wmma_f32_16x16x4_f32(
                    false, a, false, b, (short)0, acc[j], false, false);
            }
        }
        __syncthreads();
    }

    // ---- epilogue: C 16x16 layout -> row = base + r + 8*hi, col = base + l15
    const int rowb = m0 + 16 * wave + 8 * hi;
    #pragma unroll
    for (int j = 0; j < 4; ++j) {
        const int col = n0 + 16 * j + l15;
        const float bi = bias ? bias[col] : 0.0f;
        #pragma unroll
        for (int r = 0; r < 8; ++r) {
            size_t idx = (size_t)(rowb + r) * N + col;
            float v = act_f(acc[j][r] + bi, act);
            if (accum) C[idx] += v;
            else       C[idx]  = v;
        }
    }
}

// ---------------------------------------------------------------------------
// LayerNorm over last dim (in place), optional SiLU after:  h = act(LN(h)*g+b)
// one row per block
// ---------------------------------------------------------------------------
__global__ __launch_bounds__(256)
void moe_ln_act(float* __restrict__ h, const float* __restrict__ g,
                const float* __restrict__ b, int dim, int act) {
    __shared__ float s1[256], s2[256];
    const int tid = threadIdx.x;
    float* row = h + (size_t)blockIdx.x * dim;
    float sum = 0.0f, sq = 0.0f;
    for (int j = tid; j < dim; j += 256) {
        float v = row[j];
        sum += v; sq += v * v;
    }
    s1[tid] = sum; s2[tid] = sq;
    __syncthreads();
    for (int s = 128; s > 0; s >>= 1) {
        if (tid < s) { s1[tid] += s1[tid + s]; s2[tid] += s2[tid + s]; }
        __syncthreads();
    }
    float mu  = s1[0] / (float)dim;
    float var = s2[0] / (float)dim - mu * mu;
    float rs  = rsqrtf(var + 1e-5f);
    for (int j = tid; j < dim; j += 256) {
        float v = (row[j] - mu) * rs * g[j] + b[j];
        if (act == 2) v = silu_f(v);
        row[j] = v;
    }
}

// ---------------------------------------------------------------------------
// out[n,:] += gate[n,e] * LN(x[n,:] + y[n,:]; ng, nb)   (dim = 768)
// ---------------------------------------------------------------------------
__global__ __launch_bounds__(256)
void moe_residual_ln_accum(const float* __restrict__ x, const float* __restrict__ y,
                           const float* __restrict__ ng, const float* __restrict__ nb,
                           const float* __restrict__ gate, int e,
                           float* __restrict__ out) {
    __shared__ float s1[256], s2[256];
    const int n   = blockIdx.x;
    const int tid = threadIdx.x;
    const float gw = gate[n * EEXP + e];
    if (gw == 0.0f) return;
    const size_t base = (size_t)n * CDIM;
    float v[3];
    float sum = 0.0f, sq = 0.0f;
    #pragma unroll
    for (int r = 0; r < 3; ++r) {
        int j = tid + 256 * r;
        v[r] = x[base + j] + y[base + j];
        sum += v[r]; sq += v[r] * v[r];
    }
    s1[tid] = sum; s2[tid] = sq;
    __syncthreads();
    for (int s = 128; s > 0; s >>= 1) {
        if (tid < s) { s1[tid] += s1[tid + s]; s2[tid] += s2[tid + s]; }
        __syncthreads();
    }
    float mu  = s1[0] / (float)CDIM;
    float var = s2[0] / (float)CDIM - mu * mu;
    float rs  = rsqrtf(var + 1e-5f);
    #pragma unroll
    for (int r = 0; r < 3; ++r) {
        int j = tid + 256 * r;
        out[base + j] += gw * ((v[r] - mu) * rs * ng[j] + nb[j]);
    }
}

// ---------------------------------------------------------------------------
// Routing helpers
// ---------------------------------------------------------------------------
// pretf[e,j] = sum_k type_emb[TYPES[e],k] * route_w1[C+k, j] + route_b1[j]
__global__ __launch_bounds__(256)
void moe_pretf(const float* __restrict__ temb, const float* __restrict__ rw1,
               const float* __restrict__ rb1, float* __restrict__ pretf) {
    const int j = blockIdx.x * 256 + threadIdx.x;   // 0..3071
    const int e = blockIdx.y;                        // 0..5
    const int ty = (e == 0) ? 0 : ((e < 4) ? 1 : 2);
    const float* t = temb + ty * (2 * CDIM);
    float s = rb1[j];
    for (int k = 0; k < 2 * CDIM; ++k)
        s += t[k] * rw1[(size_t)(CDIM + k) * HID + j];
    pretf[(size_t)e * HID + j] = s;
}

// w2m[j] = mean_e route_w2[j,e]
__global__ void moe_w2mean(const float* __restrict__ rw2, float* __restrict__ w2m) {
    int j = blockIdx.x * 256 + threadIdx.x;
    if (j >= HID) return;
    float s = 0.0f;
    #pragma unroll
    for (int e = 0; e < EEXP; ++e) s += rw2[(size_t)j * EEXP + e];
    w2m[j] = s * (1.0f / EEXP);
}

// logits[n,e] = sum_j gelu(xr[n,j] + pretf[e,j]) * w2m[j] + mean(route_b2)
__global__ __launch_bounds__(256)
void moe_logits(const float* __restrict__ xr, const float* __restrict__ pretf,
                const float* __restrict__ w2m, const float* __restrict__ rb2,
                float* __restrict__ logits) {
    __shared__ float red[256];
    const int n   = blockIdx.x;
    const int tid = threadIdx.x;
    float acc[EEXP] = {0, 0, 0, 0, 0, 0};
    const float* xrow = xr + (size_t)n * HID;
    for (int j = tid; j < HID; j += 256) {
        float xv = xrow[j];
        float wm = w2m[j];
        #pragma unroll
        for (int e = 0; e < EEXP; ++e)
            acc[e] += gelu_f(xv + pretf[(size_t)e * HID + j]) * wm;
    }
    float b2m = (rb2[0] + rb2[1] + rb2[2] + rb2[3] + rb2[4] + rb2[5]) * (1.0f / EEXP);
    for (int e = 0; e < EEXP; ++e) {
        red[tid] = acc[e];
        __syncthreads();
        for (int s = 128; s > 0; s >>= 1) {
            if (tid < s) red[tid] += red[tid + s];
            __syncthreads();
        }
        if (tid == 0) logits[(size_t)n * EEXP + e] = red[0] + b2m;
        __syncthreads();
    }
}

// z1[n,i] = gelu( x[n,:] @ noise_w1[:,i] + noise_b1[i] ),  i < 12
__global__ void moe_noise_z1(const float* __restrict__ x, const float* __restrict__ nw1,
                             const float* __restrict__ nb1, float* __restrict__ z1) {
    int idx = blockIdx.x * 256 + threadIdx.x;
    if (idx >= NTOK * 2 * EEXP) return;
    int n = idx / (2 * EEXP);
    int i = idx % (2 * EEXP);
    const float* xr = x + (size_t)n * CDIM;
    float s = nb1[i];
    for (int k = 0; k < CDIM; ++k) s += xr[k] * nw1[(size_t)k * (2 * EEXP) + i];
    z1[idx] = gelu_f(s);
}

// gating: z2 -> softplus(softplus) -> noisy logits -> top2 -> softmax scatter
__global__ void moe_gating(const float* __restrict__ z1, const float* __restrict__ nw2,
                           const float* __restrict__ nb2, const float* __restrict__ logits,
                           const float* __restrict__ noise, const float* __restrict__ temp_p,
                           float* __restrict__ gate) {
    int n = blockIdx.x * 256 + threadIdx.x;
    if (n >= NTOK) return;
    float z2[EEXP];
    #pragma unroll
    for (int e = 0; e < EEXP; ++e) z2[e] = nb2[e];
    #pragma unroll
    for (int i = 0; i < 2 * EEXP; ++i) {
        float v = z1[(size_t)n * (2 * EEXP) + i];
        #pragma unroll
        for (int e = 0; e < EEXP; ++e) z2[e] += v * nw2[i * EEXP + e];
    }
    float temp = temp_p[0] * powf(0.95f, 10.0f);   // T=1024 -> T//100 = 10
    temp = fminf(fmaxf(temp, 0.05f), 3.0f);
    float nl[EEXP];
    #pragma unroll
    for (int e = 0; e < EEXP; ++e) {
        float ns = softplus_f(softplus_f(z2[e]));
        float bonus = (e >= 1 && e <= 3) ? 0.3f : 0.0f;   // wide experts
        nl[e] = logits[(size_t)n * EEXP + e] + temp * noise[(size_t)n * EEXP + e] * ns + bonus;
    }
    int i0 = 0;
    #pragma unroll
    for (int e = 1; e < EEXP; ++e) if (nl[e] > nl[i0]) i0 = e;
    int i1 = (i0 == 0) ? 1 : 0;
    #pragma unroll
    for (int e = 0; e < EEXP; ++e) if (e != i0 && nl[e] > nl[i1]) i1 = e;
    float e0 = 1.0f;                         // exp(v0 - v0)
    float e1 = expf(nl[i1] - nl[i0]);
    float inv = 1.0f / (e0 + e1);
    #pragma unroll
    for (int e = 0; e < EEXP; ++e) gate[(size_t)n * EEXP + e] = 0.0f;
    gate[(size_t)n * EEXP + i0] = e0 * inv;
    gate[(size_t)n * EEXP + i1] = e1 * inv;
}

__global__ void moe_zero(float* __restrict__ p, int nelem) {
    int i = blockIdx.x * 256 + threadIdx.x;
    if (i < nelem) p[i] = 0.0f;
}

// ---------------------------------------------------------------------------
// Host-side orchestration
// ---------------------------------------------------------------------------
extern "C" void kernel_launch(void* const* d_in, const int* in_sizes, int n_in,
                              void* d_out, int out_size, void* d_ws, size_t ws_size,
                              hipStream_t stream) {
    auto F = [&](int i) { return (const float*)d_in[i]; };

    const float* x        = F(0);   // [8192, 768]
    const float* noise    = F(1);   // [8192, 6]
    const float* temb     = F(2);   // [3, 1536]
    const float* rw1      = F(3);   // [2304, 3072]
    const float* rb1      = F(4);   // [3072]
    const float* rw2      = F(5);   // [3072, 6]
    const float* rb2      = F(6);   // [6]
    const float* nw1      = F(7);   // [768, 12]
    const float* nb1      = F(8);   // [12]
    const float* nw2      = F(9);   // [12, 6]
    const float* nb2      = F(10);  // [6]
    const float* temp     = F(11);  // [1]

    // workspace layout (floats)
    float* ws = (float*)d_ws;
    const size_t H1_OFF  = 0;                             // [8192,3072]
    const size_t H2_OFF  = H1_OFF + (size_t)NTOK * HID;   // [8192,3072]
    const size_t Y_OFF   = H2_OFF + (size_t)NTOK * HID;   // [8192,768]
    const size_t PT_OFF  = Y_OFF  + (size_t)NTOK * CDIM;  // [6,3072]
    const size_t W2M_OFF = PT_OFF + (size_t)EEXP * HID;
    const size_t LOG_OFF = W2M_OFF + HID;
    const size_t Z1_OFF  = LOG_OFF + (size_t)NTOK * EEXP;
    const size_t G_OFF   = Z1_OFF  + (size_t)NTOK * 2 * EEXP;
    float* H1 = ws + H1_OFF;
    float* H2 = ws + H2_OFF;
    float* Y  = ws + Y_OFF;
    float* PT = ws + PT_OFF;
    float* WM = ws + W2M_OFF;
    float* LG = ws + LOG_OFF;
    float* Z1 = ws + Z1_OFF;
    float* G  = ws + G_OFF;
    float* out = (float*)d_out;

    const dim3 blk128(128), blk256(256);
    auto gemm = [&](const float* A, const float* W, const float* bias, float* C,
                    int M, int N, int K, int act, int accum) {
        dim3 grid(N / 64, M / 64);
        moe_wmma_gemm<<<grid, blk128, 0, stream>>>(A, W, bias, C, M, N, K, act, accum);
    };

    // ---------------- routing ----------------
    moe_pretf<<<dim3(HID / 256, EEXP), blk256, 0, stream>>>(temb, rw1, rb1, PT);
    moe_w2mean<<<dim3((HID + 255) / 256), blk256, 0, stream>>>(rw2, WM);
    gemm(x, rw1, nullptr, H1, NTOK, HID, CDIM, /*act*/0, 0);   // xr = x @ rw1[:C]
    moe_logits<<<dim3(NTOK), blk256, 0, stream>>>(H1, PT, WM, rb2, LG);
    moe_noise_z1<<<dim3((NTOK * 2 * EEXP + 255) / 256), blk256, 0, stream>>>(x, nw1, nb1, Z1);
    moe_gating<<<dim3((NTOK + 255) / 256), blk256, 0, stream>>>(Z1, nw2, nb2, LG, noise, temp, G);

    // ---------------- output init ----------------
    moe_zero<<<dim3((NTOK * CDIM + 255) / 256), blk256, 0, stream>>>(out, NTOK * CDIM);

    // ---------------- experts ----------------
    int idx = 12;   // expert params start (dict insertion order)
    const int TYPES[EEXP] = {0, 1, 1, 1, 2, 2};
    for (int e = 0; e < EEXP; ++e) {
        if (TYPES[e] == 0) {
            const float* w1 = F(idx + 0), *b1 = F(idx + 1);
            const float* w2 = F(idx + 2), *b2 = F(idx + 3);
            const float* lg = F(idx + 4), *lb = F(idx + 5);
            const float* w3 = F(idx + 6), *b3 = F(idx + 7);
            const float* ng = F(idx + 8), *nb = F(idx + 9);
            idx += 10;
            gemm(x,  w1, b1, H1, NTOK, HID, CDIM, /*silu*/2, 0);
            gemm(H1, w2, b2, H2, NTOK, HID, HID,  /*none*/0, 0);
            moe_ln_act<<<dim3(NTOK), blk256, 0, stream>>>(H2, lg, lb, HID, /*silu*/2);
            gemm(H2, w3, b3, Y, NTOK, CDIM, HID, 0, 0);
            moe_residual_ln_accum<<<dim3(NTOK), blk256, 0, stream>>>(x, Y, ng, nb, G, e, out);
        } else if (TYPES[e] == 1) {
            const float* w1 = F(idx + 0), *b1 = F(idx + 1);
            const float* lg = F(idx + 2), *lb = F(idx + 3);
            const float* w2 = F(idx + 4), *b2 = F(idx + 5);
            const float* ng = F(idx + 6), *nb = F(idx + 7);
            idx += 8;
            gemm(x, w1, b1, H1, NTOK, HID, CDIM, /*gelu*/1, 0);
            moe_ln_act<<<dim3(NTOK), blk256, 0, stream>>>(H1, lg, lb, HID, /*none*/0);
            gemm(H1, w2, b2, Y, NTOK, CDIM, HID, 0, 0);
            moe_residual_ln_accum<<<dim3(NTOK), blk256, 0, stream>>>(x, Y, ng, nb, G, e, out);
        } else {
            const float* a1 = F(idx + 0), *ab1 = F(idx + 1);
            const float* a2 = F(idx + 2), *ab2 = F(idx + 3);
            const float* c1 = F(idx + 4), *cb1 = F(idx + 5);
            const float* c2 = F(idx + 6), *cb2 = F(idx + 7);
            const float* pj = F(idx + 8), *pjb = F(idx + 9);
            const float* ng = F(idx + 10), *nb = F(idx + 11);
            idx += 12;
            float* P1 = H2;
            float* P2 = H2 + (size_t)NTOK * CDIM;
            gemm(x,  a1, ab1, H1, NTOK, HID, CDIM, /*gelu*/1, 0);
            gemm(H1, a2, ab2, P1, NTOK, CDIM, HID, 0, 0);
            gemm(x,  c1, cb1, H1, NTOK, HALF_H, CDIM, /*silu*/2, 0);
            gemm(H1, c2, cb2, P2, NTOK, CDIM, HALF_H, 0, 0);
            gemm(P1, pj, pjb, Y, NTOK, CDIM, CDIM, 0, 0);                           // proj[:C]
            gemm(P2, pj + (size_t)CDIM * CDIM, nullptr, Y, NTOK, CDIM, CDIM, 0, 1); // proj[C:]
            moe_residual_ln_accum<<<dim3(NTOK), blk256, 0, stream>>>(x, Y, ng, nb, G, e, out);
        }
    }
    (void)in_sizes; (void)n_in; (void)out_size; (void)ws_size;
}